// _PositionalEncoding_60395830117137
// MI455X (gfx1250) — compile-verified
//
#include <hip/hip_runtime.h>
#include <math.h>

// out[b,s,d] = x[b,s,d] + pe[d], pe[d] constant per d (same for all positions).
// x: [8, 4096, 1024] f32. Pure streaming: 268 MB total -> ~11.5 us at 23.3 TB/s.
//
// 256 threads/block (8 wave32s); thread t owns dims 4t..4t+3 -> its 4 PE values
// are loop-invariant VGPRs. PE prologue uses a compact Cody-Waite sin/cos valid
// for |x| <= 2^15 (our args <= 9816), avoiding libm's Payne-Hanek slow path.
// Each block streams 32 rows: outer loop x4 (not unrolled) over an unrolled
// 8-row body: 8 b128 NT loads in flight, then 8 add + b128 NT stores.

typedef __attribute__((ext_vector_type(4))) float v4f;

#define EMB 1024
#define VEC 4
#define THREADS (EMB / VEC)      // 256 threads = one full row per iteration
#define INNER_ROWS 8
#define OUTER_ITERS 4
#define ROWS_PER_BLOCK (INNER_ROWS * OUTER_ITERS)   // 32

// sin & cos for 0 <= x < 32768, straight-line (no branches, no slow path).
__device__ __forceinline__ void sincos_small(float x, float* s, float* c) {
    const float INV_PI = 0.31830988618367195f;   // 1/pi
    const float PI_HI  = 3.14159274101257324f;   // float(pi)
    const float PI_LO  = -8.74227765734758577e-8f; // pi - PI_HI
    float nf = rintf(x * INV_PI);
    float r  = fmaf(nf, -PI_HI, x);
    r        = fmaf(nf, -PI_LO, r);              // r in ~[-pi/2, pi/2]
    const float r2 = r * r;
    // sin(r): r*(1 - r2/6 + r2^2/120 - r2^3/5040 + r2^4/362880 - r2^5/39916800)
    float sp = fmaf(r2, -2.50521083854417188e-8f, 2.75573192239858907e-6f);
    sp = fmaf(r2, sp, -1.98412698412698413e-4f);
    sp = fmaf(r2, sp,  8.33333333333333322e-3f);
    sp = fmaf(r2, sp, -1.66666666666666666e-1f);
    float sv = r * fmaf(r2, sp, 1.0f);
    // cos(r): 1 - r2/2 + r2^2/24 - r2^3/720 + r2^4/40320 - r2^5/3628800 + r2^6/479001600
    float cp = fmaf(r2,  2.08767569878680990e-9f, -2.75573192239858883e-7f);
    cp = fmaf(r2, cp,  2.48015873015873016e-5f);
    cp = fmaf(r2, cp, -1.38888888888888894e-3f);
    cp = fmaf(r2, cp,  4.16666666666666666e-2f);
    cp = fmaf(r2, cp, -5.00000000000000000e-1f);
    float cv = fmaf(r2, cp, 1.0f);
    // sin(x) = (-1)^n sin(r), cos(x) = (-1)^n cos(r)
    const float sgn = ((int)nf & 1) ? -1.0f : 1.0f;
    *s = sv * sgn;
    *c = cv * sgn;
}

__global__ __launch_bounds__(THREADS)
void _PositionalEncoding_pe_add_kernel(const float* __restrict__ x,
                                       float* __restrict__ out,
                                       int nrows) {
    const int t  = threadIdx.x;      // 0..255
    const int d0 = t * VEC;          // even; first of this thread's 4 dims

    // Reference: den_i = 10000^(2i/1024); pe[2i]=sin(den_i); pe[2i+1]=cos(den_i)
    const float K = 13.287712379549449f / (float)EMB;   // log2(10000)/1024
    const float den0 = exp2f((float)(d0)     * K);      // <= 9816 < 2^15
    const float den1 = exp2f((float)(d0 + 2) * K);
    float s0, c0, s1, c1;
    sincos_small(den0, &s0, &c0);
    sincos_small(den1, &s1, &c1);
    v4f pe;
    pe[0] = s0; pe[1] = c0; pe[2] = s1; pe[3] = c1;

    const long long row0 = (long long)blockIdx.x * ROWS_PER_BLOCK;
    const v4f* __restrict__ src = (const v4f*)x + row0 * THREADS + t;
    v4f*       __restrict__ dst = (v4f*)out     + row0 * THREADS + t;

    if (row0 + ROWS_PER_BLOCK <= (long long)nrows) {
        #pragma unroll 1
        for (int o = 0; o < OUTER_ITERS; ++o) {
            const long long base = (long long)o * INNER_ROWS * THREADS;
            v4f v[INNER_ROWS];
            #pragma unroll
            for (int r = 0; r < INNER_ROWS; ++r)
                v[r] = __builtin_nontemporal_load(src + base + (long long)r * THREADS);
            #pragma unroll
            for (int r = 0; r < INNER_ROWS; ++r) {
                v4f s = v[r] + pe;
                __builtin_nontemporal_store(s, dst + base + (long long)r * THREADS);
            }
        }
    } else {
        // Tail path (not taken for the reference shape, kept for safety).
        #pragma unroll 1
        for (int r = 0; r < ROWS_PER_BLOCK; ++r) {
            if (row0 + r < (long long)nrows) {
                v4f s = __builtin_nontemporal_load(src + (long long)r * THREADS) + pe;
                __builtin_nontemporal_store(s, dst + (long long)r * THREADS);
            }
        }
    }
}

extern "C" void kernel_launch(void* const* d_in, const int* in_sizes, int n_in,
                              void* d_out, int out_size, void* d_ws, size_t ws_size,
                              hipStream_t stream) {
    (void)n_in; (void)d_ws; (void)ws_size; (void)out_size;
    const float* x = (const float*)d_in[0];
    float* out     = (float*)d_out;

    const int total = in_sizes[0];          // 8*4096*1024
    const int nrows = total / EMB;          // 32768 rows of 1024 floats

    const int grid = (nrows + ROWS_PER_BLOCK - 1) / ROWS_PER_BLOCK;  // 1024
    _PositionalEncoding_pe_add_kernel<<<grid, THREADS, 0, stream>>>(x, out, nrows);
}